// Top1Gate_44212393345663
// MI455X (gfx1250) — compile-verified
//
#include <hip/hip_runtime.h>
#include <math.h>

typedef __attribute__((ext_vector_type(2))) float v2f;
typedef __attribute__((ext_vector_type(8))) float v8f;

#define MODEL_DIM   4096
#define NUM_EXPERTS 64
#define NUM_TOKENS  8192
#define KC          64          // K-chunk staged in LDS
#define NCHUNK      (MODEL_DIM / KC)
#define ROW_STRIDE  68          // padded LDS row stride (floats): 68%64==4 -> conflict-free WMMA reads
#define LOG_STRIDE  68          // padded logits stride
#define TOK_PER_BLOCK 64
#define TOK_PER_WAVE  16
#define BUF_FLOATS  (TOK_PER_BLOCK * ROW_STRIDE)   // 4352 floats = 17408 B per tile buffer

// Output layout (floats): [0,8192) indices | [8192] capacity | [8193,16385) locations
//                         | [16385,24577) gates | [24577] num_experts
#define OFF_IDX  0
#define OFF_CAP  8192
#define OFF_LOC  8193
#define OFF_GATE 16385
#define OFF_NE   24577

__global__ __launch_bounds__(128) void top1_gate_logits_kernel(
    const float* __restrict__ x, const float* __restrict__ W,
    float* __restrict__ out, int* __restrict__ idx_ws)
{
    __shared__ float XL[2][BUF_FLOATS];   // double-buffered 64 tokens x KC
    __shared__ float WL[2][BUF_FLOATS];   // double-buffered 64 experts x KC
    // logits scratch aliases XL[0] (4*16*LOG_STRIDE == 4352 floats, used only after the k-loop)

    const int tid  = threadIdx.x;
    const int wave = tid >> 5;
    const int lane = tid & 31;
    const int r16  = lane & 15;   // row-within-16 (M for A, N for B)
    const int h    = lane >> 4;   // half-wave selects K pair {0,1} vs {2,3}
    const int tok0 = blockIdx.x * TOK_PER_BLOCK;

    // LDS byte offsets (generic shared pointer low 32 bits == LDS offset)
    const unsigned xl_base = (unsigned)(uintptr_t)(&XL[0][0]);
    const unsigned wl_base = (unsigned)(uintptr_t)(&WL[0][0]);

    // staging coordinates for this thread (8 x 16B per chunk, coalesced)
    const int flat0 = tid * 4;            // float offset of this thread's first float4 slot

    // Issue one K-chunk's async fills into buffer `buf` (16 async instructions per wave).
    auto issue_chunk = [&](int k0, int buf) {
        #pragma unroll
        for (int i = 0; i < (TOK_PER_BLOCK * KC / 4) / 128; ++i) {  // 8 iterations
            int flat = i * 512 + flat0;   // 128 threads * 4 floats
            int row  = flat >> 6;         // /KC
            int col  = flat & (KC - 1);
            unsigned lds_x = xl_base + (unsigned)buf * (BUF_FLOATS * 4u)
                                     + 4u * (unsigned)(row * ROW_STRIDE + col);
            unsigned lds_w = wl_base + (unsigned)buf * (BUF_FLOATS * 4u)
                                     + 4u * (unsigned)(row * ROW_STRIDE + col);
            unsigned long long gx =
                (unsigned long long)(uintptr_t)(x + (size_t)(tok0 + row) * MODEL_DIM + k0 + col);
            unsigned long long gw =
                (unsigned long long)(uintptr_t)(W + (size_t)row * MODEL_DIM + k0 + col);
            asm volatile("global_load_async_to_lds_b128 %0, %1, off"
                         :: "v"(lds_x), "v"(gx) : "memory");
            asm volatile("global_load_async_to_lds_b128 %0, %1, off"
                         :: "v"(lds_w), "v"(gw) : "memory");
        }
    };

    v8f acc0 = {}, acc1 = {}, acc2 = {}, acc3 = {};

    issue_chunk(0, 0);   // prologue: fill buffer 0 with chunk 0

    for (int c = 0; c < NCHUNK; ++c) {
        const int cur = c & 1;
        const int k0  = c * KC;

        if (c + 1 < NCHUNK) {
            // buffer cur^1 was fully consumed at iteration c-1 (end barrier) -> safe to refill
            issue_chunk(k0 + KC, cur ^ 1);
            // warm L2 one chunk further ahead (gfx1250 global_prefetch_b8)
            if (c + 2 < NCHUNK) {
                __builtin_prefetch(x + (size_t)(tok0 + (tid >> 1)) * MODEL_DIM + k0 + 2 * KC + (tid & 1) * 32, 0, 3);
                __builtin_prefetch(W + (size_t)(tid >> 1) * MODEL_DIM + k0 + 2 * KC + (tid & 1) * 32, 0, 3);
            }
            // asyncs retire in order: draining to <=16 outstanding means chunk c's 16 fills landed
            asm volatile("s_wait_asynccnt 0x10" ::: "memory");
        } else {
            asm volatile("s_wait_asynccnt 0x0" ::: "memory");
        }
        __syncthreads();   // publish buffer `cur` to all waves

        // ---- WMMA: 16 tokens x 64 experts, K in steps of 4 ----
        const float* Xc = &XL[cur][0];
        const float* Wc = &WL[cur][0];
        const int arow = wave * TOK_PER_WAVE + r16;
        #pragma unroll
        for (int kk = 0; kk < KC; kk += 4) {
            // A (16x4 f32): lane holds x[m = r16][k = kk + 2h + {0,1}]
            v2f a  = *reinterpret_cast<const v2f*>(&Xc[arow * ROW_STRIDE + kk + 2 * h]);
            // B (4x16 f32): lane holds W[n = 16g + r16][k = kk + 2h + {0,1}]
            v2f b0 = *reinterpret_cast<const v2f*>(&Wc[(r16 +  0) * ROW_STRIDE + kk + 2 * h]);
            v2f b1 = *reinterpret_cast<const v2f*>(&Wc[(r16 + 16) * ROW_STRIDE + kk + 2 * h]);
            v2f b2 = *reinterpret_cast<const v2f*>(&Wc[(r16 + 32) * ROW_STRIDE + kk + 2 * h]);
            v2f b3 = *reinterpret_cast<const v2f*>(&Wc[(r16 + 48) * ROW_STRIDE + kk + 2 * h]);
            acc0 = __builtin_amdgcn_wmma_f32_16x16x4_f32(false, a, false, b0, (short)0, acc0, false, false);
            acc1 = __builtin_amdgcn_wmma_f32_16x16x4_f32(false, a, false, b1, (short)0, acc1, false, false);
            acc2 = __builtin_amdgcn_wmma_f32_16x16x4_f32(false, a, false, b2, (short)0, acc2, false, false);
            acc3 = __builtin_amdgcn_wmma_f32_16x16x4_f32(false, a, false, b3, (short)0, acc3, false, false);
        }
        __syncthreads();   // all waves done reading buffer `cur` -> next iter may refill it
    }

    // ---- spill logits tile (16x64) to padded LDS (aliases XL[0]): M = j + 8h, N = 16g + r16 ----
    float* LOG = &XL[0][0];
    float* lw  = &LOG[wave * TOK_PER_WAVE * LOG_STRIDE];
    #pragma unroll
    for (int j = 0; j < 8; ++j) {
        int M = j + 8 * h;
        lw[M * LOG_STRIDE + r16 +  0] = acc0[j];
        lw[M * LOG_STRIDE + r16 + 16] = acc1[j];
        lw[M * LOG_STRIDE + r16 + 32] = acc2[j];
        lw[M * LOG_STRIDE + r16 + 48] = acc3[j];
    }
    __syncthreads();   // cross-lane LDS dependency within/among waves

    // ---- per-token argmax + softmax gate (lanes 0..15, one token each) ----
    if (lane < 16) {
        const float* row = &lw[lane * LOG_STRIDE];
        float m = row[0];
        int   am = 0;
        #pragma unroll 8
        for (int e = 1; e < NUM_EXPERTS; ++e) {
            float v = row[e];
            if (v > m) { m = v; am = e; }      // strict '>' keeps first occurrence (jnp.argmax)
        }
        float s = 0.0f;
        #pragma unroll 8
        for (int e = 0; e < NUM_EXPERTS; ++e) s += __expf(row[e] - m);
        float gate = 1.0f / s;                 // softmax value at the argmax = exp(0)/sum

        int token = tok0 + wave * TOK_PER_WAVE + lane;
        out[OFF_IDX  + token] = (float)am;
        out[OFF_GATE + token] = gate;
        idx_ws[token] = am;
    }
}

// One wave per expert: order-preserving exclusive count via wave32 ballot prefix.
__global__ __launch_bounds__(32) void top1_gate_locations_kernel(
    const int* __restrict__ idx, float* __restrict__ out)
{
    const int e    = blockIdx.x;
    const int lane = threadIdx.x;
    int running = 0;
    for (int base = 0; base < NUM_TOKENS; base += 32) {
        int t = base + lane;
        bool match = (idx[t] == e);
        unsigned mask = __builtin_amdgcn_ballot_w32(match);
        if (match) {
            int rank = __popc(mask & ((1u << lane) - 1u));
            out[OFF_LOC + t] = (float)(running + rank);
        }
        running += __popc(mask);
    }
    if (e == 0 && lane == 0) {
        out[OFF_CAP] = 128.0f;   // int(1.0 * ceil(8192/64))
        out[OFF_NE]  = 64.0f;    // num_experts
    }
}

extern "C" void kernel_launch(void* const* d_in, const int* in_sizes, int n_in,
                              void* d_out, int out_size, void* d_ws, size_t ws_size,
                              hipStream_t stream) {
    const float* x = (const float*)d_in[0];   // [8192, 4096] fp32
    const float* W = (const float*)d_in[1];   // [64, 4096] fp32
    float* out     = (float*)d_out;           // 24578 floats (tuple concatenated)
    int*   idx_ws  = (int*)d_ws;              // 8192 int32 scratch

    top1_gate_logits_kernel<<<NUM_TOKENS / TOK_PER_BLOCK, 128, 0, stream>>>(x, W, out, idx_ws);
    top1_gate_locations_kernel<<<NUM_EXPERTS, 32, 0, stream>>>(idx_ws, out);
}